// bionet_37091337568613
// MI455X (gfx1250) — compile-verified
//
#include <hip/hip_runtime.h>
#include <cstdint>
#include <cstddef>

// Problem constants (match reference)
#define LEAK    0.01f
#define ITERS   120
#define BATCHC  128      // batch columns; one wave32 covers it with 4 floats/lane
#define WPB     8        // waves per block (blockDim = 256)
#define DEPTH   8        // async gathers per pipeline chunk
#define SLOTS   16       // LDS ring slots per wave (2 chunks in flight)

// ---------------------------------------------------------------------------
// CSR construction
// ---------------------------------------------------------------------------
__global__ void k_hist(const int* __restrict__ tgt, int* __restrict__ counts, int E) {
    int e = blockIdx.x * blockDim.x + threadIdx.x;
    if (e < E) atomicAdd(&counts[tgt[e]], 1);
}

// Single-block exclusive scan over n counts -> rowptr[0..n]
__global__ void k_scan(const int* __restrict__ counts, int* __restrict__ rowptr, int n) {
    __shared__ int tmp[1024];
    __shared__ int carry;
    int tid = threadIdx.x;
    if (tid == 0) carry = 0;
    __syncthreads();
    for (int base = 0; base < n; base += 1024) {
        int v = (base + tid < n) ? counts[base + tid] : 0;
        tmp[tid] = v;
        __syncthreads();
        for (int off = 1; off < 1024; off <<= 1) {
            int t = (tid >= off) ? tmp[tid - off] : 0;
            __syncthreads();
            tmp[tid] += t;
            __syncthreads();
        }
        int incl = tmp[tid];
        int c = carry;
        __syncthreads();                       // all reads of carry done
        if (tid == 1023) carry = c + incl;     // chunk total
        if (base + tid < n) rowptr[base + tid] = c + incl - v;  // exclusive
        __syncthreads();
    }
    if (threadIdx.x == 0) rowptr[n] = carry;
}

// Scatter into CSR; source index premultiplied by row stride (BATCHC elements)
__global__ void k_scatter(const int* __restrict__ tgt, const int* __restrict__ src,
                          const float* __restrict__ w, const int* __restrict__ rowptr,
                          int* __restrict__ fill, float* __restrict__ csr_w,
                          int* __restrict__ csr_s, int E) {
    int e = blockIdx.x * blockDim.x + threadIdx.x;
    if (e >= E) return;
    int t   = tgt[e];
    int pos = atomicAdd(&fill[t], 1);
    int idx = rowptr[t] + pos;
    csr_w[idx] = w[e];
    csr_s[idx] = src[e] * BATCHC;   // element offset of the source row
}

// bIn[t, b] = x[b, t] + bias[t]   (node-major layout for the hot loop)
__global__ void k_bin(const float* __restrict__ x, const float* __restrict__ bias,
                      float* __restrict__ bIn, int size) {
    int t = blockIdx.x;
    int b = threadIdx.x;
    bIn[(size_t)t * BATCHC + b] = x[(size_t)b * size + t] + bias[t];
}

// ---------------------------------------------------------------------------
// One fixed-point iteration: xout = leakyReLU(A*xin + bIn)
// One wave32 per output row. Per-wave 16-slot LDS ring; async global->LDS
// gathers double-buffered against the FMA consume loop. Full 8-edge chunks
// take an unrolled fast path; ragged tails take a dynamic path.
// ---------------------------------------------------------------------------
__global__ void __launch_bounds__(WPB * 32)
k_iter(const float* __restrict__ bIn, const float* __restrict__ xin,
       float* __restrict__ xout, const int* __restrict__ rowptr,
       const float* __restrict__ csr_w, const int* __restrict__ csr_s, int nrows) {
    __shared__ __align__(16) float stage[WPB][SLOTS][BATCHC];  // 64 KB

    const int wave = threadIdx.x >> 5;
    const int lane = threadIdx.x & 31;
    const int row  = blockIdx.x * WPB + wave;
    if (row >= nrows) return;

    // Force wave-uniformity so the edge stream uses the scalar path
    const int rowu = __builtin_amdgcn_readfirstlane(row);
    const int beg  = rowptr[rowu];
    const int end  = rowptr[rowu + 1];
    const int n    = end - beg;
    const float* __restrict__ wp = csr_w + beg;
    const int*   __restrict__ sp = csr_s + beg;

    // LDS byte address of this lane's 16B slice of slot 0 (AS3 offset = low 32 bits)
    const uint32_t lds0   = (uint32_t)(uintptr_t)(&stage[wave][0][lane * 4]);
    const float*   gxlane = xin + (lane << 2);

    float acc0 = 0.f, acc1 = 0.f, acc2 = 0.f, acc3 = 0.f;

    // Issue one 512B row gather (16B/lane x 32 lanes) for edge e into ring slot e&15
    auto issue = [&](int e) {
        const int sj = sp[e];                       // uniform: element offset of row
        const float* g = gxlane + sj;
        const uint32_t l = lds0 + (uint32_t)((e & (SLOTS - 1)) * (BATCHC * 4));
        asm volatile("global_load_async_to_lds_b128 %0, %1, off"
                     :: "v"(l), "v"(g) : "memory");
    };
    auto consume = [&](int e) {
        const float wj = wp[e];                     // uniform weight
        const float4 xr = *(const float4*)&stage[wave][e & (SLOTS - 1)][lane * 4];
        acc0 = fmaf(wj, xr.x, acc0);
        acc1 = fmaf(wj, xr.y, acc1);
        acc2 = fmaf(wj, xr.z, acc2);
        acc3 = fmaf(wj, xr.w, acc3);
    };

    // Prologue: issue chunk 0
    if (n >= DEPTH) {
#pragma unroll
        for (int j = 0; j < DEPTH; ++j) issue(j);
    } else {
        for (int j = 0; j < n; ++j) issue(j);
    }

    // Steady state: current chunk is full
    int base = 0;
    while (base + DEPTH <= n) {
        const int nbase = base + DEPTH;
        if (nbase + DEPTH <= n) {
            // Next chunk full: issue unrolled, then wait for current (<=8 in flight,
            // async loads complete in order => current chunk has landed)
#pragma unroll
            for (int j = 0; j < DEPTH; ++j) issue(nbase + j);
            asm volatile("s_wait_asynccnt 8" ::: "memory");
        } else {
            // Ragged/empty next chunk: issue it, then drain everything
            const int nq = n - nbase;               // 0..7
            for (int j = 0; j < nq; ++j) issue(nbase + j);
            asm volatile("s_wait_asynccnt 0" ::: "memory");
        }
#pragma unroll
        for (int j = 0; j < DEPTH; ++j) consume(base + j);
        base = nbase;
    }

    // Tail (already issued; counter already drained unless n < DEPTH)
    if (base < n) {
        asm volatile("s_wait_asynccnt 0" ::: "memory");
        for (int j = base; j < n; ++j) consume(j);
    }

    // Epilogue: add bIn, leakyReLU, store contiguous 512B row
    const size_t o = (size_t)row * BATCHC + lane * 4;
    const float4 bv = *(const float4*)&bIn[o];
    float v0 = acc0 + bv.x, v1 = acc1 + bv.y, v2 = acc2 + bv.z, v3 = acc3 + bv.w;
    v0 = (v0 < 0.f) ? LEAK * v0 : v0;
    v1 = (v1 < 0.f) ? LEAK * v1 : v1;
    v2 = (v2 < 0.f) ? LEAK * v2 : v2;
    v3 = (v3 < 0.f) ? LEAK * v3 : v3;
    float4 ov; ov.x = v0; ov.y = v1; ov.z = v2; ov.w = v3;
    *(float4*)&xout[o] = ov;
}

// ---------------------------------------------------------------------------
// Final transpose: out[b, t] = xhat[t, b]   (tiled via LDS)
// ---------------------------------------------------------------------------
__global__ void k_tr(const float* __restrict__ xhat, float* __restrict__ out, int size) {
    __shared__ float tile[32][33];
    const int tt = blockIdx.x * 32;   // node tile
    const int tb = blockIdx.y * 32;   // batch tile
    const int lx = threadIdx.x;       // 0..31
    const int ly = threadIdx.y;       // 0..7
    for (int i = ly; i < 32; i += 8) {
        if (tt + i < size)
            tile[i][lx] = xhat[(size_t)(tt + i) * BATCHC + (tb + lx)];
    }
    __syncthreads();
    for (int i = ly; i < 32; i += 8) {
        if (tt + lx < size)
            out[(size_t)(tb + i) * size + (tt + lx)] = tile[lx][i];
    }
}

// ---------------------------------------------------------------------------
extern "C" void kernel_launch(void* const* d_in, const int* in_sizes, int n_in,
                              void* d_out, int out_size, void* d_ws, size_t ws_size,
                              hipStream_t stream) {
    const float* x      = (const float*)d_in[0];  // [B, S]
    const float* wts    = (const float*)d_in[1];  // [E]
    const float* bias   = (const float*)d_in[2];  // [S]
    const int*   etgt   = (const int*)d_in[3];    // [E]
    const int*   esrc   = (const int*)d_in[4];    // [E]

    const int E = in_sizes[1];
    const int S = in_sizes[2];

    // Workspace layout (512B aligned segments); total ~36 MB
    uint8_t* w8 = (uint8_t*)d_ws;
    size_t off = 0;
    auto take = [&](size_t bytes) -> void* {
        off = (off + 511) & ~(size_t)511;
        void* p = w8 + off;
        off += bytes;
        return p;
    };
    float* bIn    = (float*)take((size_t)S * BATCHC * sizeof(float));
    float* xA     = (float*)take((size_t)S * BATCHC * sizeof(float));
    float* xB     = (float*)take((size_t)S * BATCHC * sizeof(float));
    float* csr_w  = (float*)take((size_t)E * sizeof(float));
    int*   csr_s  = (int*)  take((size_t)E * sizeof(int));
    int*   rowptr = (int*)  take((size_t)(S + 1) * sizeof(int));
    int*   counts = (int*)  take((size_t)S * sizeof(int));
    (void)ws_size;

    // 1) CSR build
    hipMemsetAsync(counts, 0, (size_t)S * sizeof(int), stream);
    k_hist<<<(E + 255) / 256, 256, 0, stream>>>(etgt, counts, E);
    k_scan<<<1, 1024, 0, stream>>>(counts, rowptr, S);
    hipMemsetAsync(counts, 0, (size_t)S * sizeof(int), stream);  // reuse as fill cursor
    k_scatter<<<(E + 255) / 256, 256, 0, stream>>>(etgt, esrc, wts, rowptr, counts,
                                                   csr_w, csr_s, E);

    // 2) bIn = x^T + bias
    k_bin<<<S, BATCHC, 0, stream>>>(x, bias, bIn, S);

    // 3) 120 ping-pong fixed-point iterations, starting from xhat = 0
    hipMemsetAsync(xA, 0, (size_t)S * BATCHC * sizeof(float), stream);
    const int iter_blocks = (S + WPB - 1) / WPB;
    float* pin  = xA;
    float* pout = xB;
    for (int i = 0; i < ITERS; ++i) {
        k_iter<<<iter_blocks, WPB * 32, 0, stream>>>(bIn, pin, pout, rowptr,
                                                     csr_w, csr_s, S);
        float* t = pin; pin = pout; pout = t;
    }

    // 4) transpose result ([S,B] -> [B,S]) into d_out
    dim3 tb(32, 8);
    dim3 tg((S + 31) / 32, BATCHC / 32);
    k_tr<<<tg, tb, 0, stream>>>(pin, (float*)d_out, S);
    (void)n_in; (void)out_size;
}